// QuantumFeatureExtractor_34548716929767
// MI455X (gfx1250) — compile-verified
//
#include <hip/hip_runtime.h>
#include <hip/hip_bf16.h>

typedef __attribute__((ext_vector_type(2))) float v2f;
typedef __attribute__((ext_vector_type(8))) float v8f;

// D = A(16x4, f32) * B(4x16, f32) + C(16x16, f32)  -- V_WMMA_F32_16X16X4_F32
__device__ __forceinline__ v8f wmma4(v2f a, v2f b, v8f c) {
    return __builtin_amdgcn_wmma_f32_16x16x4_f32(false, a, false, b, (short)0, c, false, false);
}

// ---------------------------------------------------------------------------
// K0: build the 96x16 real matrix A = [ReU1;ImU1;ReU2;ImU2;ReU3;ImU3]
// 48 threads: thread (c,col) evolves basis state |col> through circuit c.
// ---------------------------------------------------------------------------
__device__ void apply1q(float* sr, float* si, int w,
                        float g00r, float g00i, float g01r, float g01i,
                        float g10r, float g10i, float g11r, float g11i) {
    int s = 8 >> w;
    for (int k = 0; k < 16; ++k) {
        if (k & s) continue;
        int k1 = k | s;
        float a0r = sr[k], a0i = si[k], a1r = sr[k1], a1i = si[k1];
        sr[k]  = g00r * a0r - g00i * a0i + g01r * a1r - g01i * a1i;
        si[k]  = g00r * a0i + g00i * a0r + g01r * a1i + g01i * a1r;
        sr[k1] = g10r * a0r - g10i * a0i + g11r * a1r - g11i * a1i;
        si[k1] = g10r * a0i + g10i * a0r + g11r * a1i + g11i * a1r;
    }
}

__global__ void k_build(const float* __restrict__ p1, const float* __restrict__ p2,
                        const float* __restrict__ p3, float* __restrict__ Amat) {
    int tid = threadIdx.x;
    if (tid >= 48) return;
    int c = tid >> 4, col = tid & 15;
    const float* P = (c == 0) ? p1 : (c == 1) ? p2 : p3;
    float sr[16], si[16];
    for (int i = 0; i < 16; ++i) { sr[i] = (i == col) ? 1.f : 0.f; si[i] = 0.f; }
    const float H = 0.70710678118654752f;
    for (int layer = 0; layer < 2; ++layer) {
        const float* pl = P + layer * 12;
        for (int w = 0; w < 4; ++w)                    // Hadamard on all wires
            apply1q(sr, si, w, H, 0.f, H, 0.f, H, 0.f, -H, 0.f);
        for (int w = 0; w < 4; ++w) {                  // RY, RZ, RX per wire
            float th = 0.5f * pl[w];
            float cs = cosf(th), sn = sinf(th);
            apply1q(sr, si, w, cs, 0.f, -sn, 0.f, sn, 0.f, cs, 0.f);
            th = 0.5f * pl[w + 4]; cs = cosf(th); sn = sinf(th);
            apply1q(sr, si, w, cs, -sn, 0.f, 0.f, 0.f, 0.f, cs, sn);
            th = 0.5f * pl[w + 8]; cs = cosf(th); sn = sinf(th);
            apply1q(sr, si, w, cs, 0.f, 0.f, -sn, 0.f, -sn, cs, 0.f);
        }
        for (int w = 0; w < 4; ++w)                    // S gate
            apply1q(sr, si, w, 1.f, 0.f, 0.f, 0.f, 0.f, 0.f, 0.f, 1.f);
        for (int e = 0; e < 4; ++e) {                  // CNOT + CZ ring
            int q1 = e, q2 = (e + 1) & 3;
            int s1 = 8 >> q1, s2 = 8 >> q2;
            for (int k = 0; k < 16; ++k)
                if ((k & s1) && !(k & s2)) {
                    int k1 = k | s2;
                    float tr = sr[k]; sr[k] = sr[k1]; sr[k1] = tr;
                    float ti = si[k]; si[k] = si[k1]; si[k1] = ti;
                }
            for (int k = 0; k < 16; ++k)
                if ((k & s1) && (k & s2)) { sr[k] = -sr[k]; si[k] = -si[k]; }
        }
    }
    for (int k = 0; k < 16; ++k) {
        Amat[(32 * c + k) * 16 + col]      = sr[k];
        Amat[(32 * c + 16 + k) * 16 + col] = si[k];
    }
}

// ---------------------------------------------------------------------------
// K1: per 16-row batch tile: normalize, Y = S * A^T via 24x WMMA f32 16x16x4,
// probs = Yr^2+Yi^2, in-register Walsh-Hadamard over lanes -> q[B][12].
// ---------------------------------------------------------------------------
__global__ __launch_bounds__(256) void k_pass1(const float* __restrict__ x,
                                               const float* __restrict__ Amat,
                                               float* __restrict__ qbuf, int ntiles) {
    const int lane = threadIdx.x & 31;
    const int m = lane & 15;
    const int kh = lane >> 4;
    const int wid = blockIdx.x * (blockDim.x >> 5) + (threadIdx.x >> 5);
    const int nw = gridDim.x * (blockDim.x >> 5);

    v2f bf[6][4];                                  // hoisted A^T B-fragments
#pragma unroll
    for (int t = 0; t < 6; ++t)
#pragma unroll
        for (int ks = 0; ks < 4; ++ks)
            bf[t][ks] = *(const v2f*)(Amat + (16 * t + m) * 16 + 4 * ks + 2 * kh);

    for (int tile = wid; tile < ntiles; tile += nw) {
        const int row0 = tile << 4;
        v2f af[4];
        float ss = 0.f;
#pragma unroll
        for (int ks = 0; ks < 4; ++ks) {
            v2f a = *(const v2f*)(x + (size_t)(row0 + m) * 16 + 4 * ks + 2 * kh);
            ss += a.x * a.x + a.y * a.y;
            af[ks] = a;
        }
        ss += __shfl_xor(ss, 16);
        float inv = rsqrtf(ss);                    // s = x / ||x||
#pragma unroll
        for (int ks = 0; ks < 4; ++ks) { af[ks].x *= inv; af[ks].y *= inv; }

        v8f d[6];
#pragma unroll
        for (int t = 0; t < 6; ++t) {
            v8f acc = {};
#pragma unroll
            for (int ks = 0; ks < 4; ++ks)
                acc = wmma4(af[ks], bf[t][ks], acc);
            d[t] = acc;
        }
#pragma unroll
        for (int c = 0; c < 3; ++c) {
            float v[8];
#pragma unroll
            for (int r = 0; r < 8; ++r)
                v[r] = d[2 * c][r] * d[2 * c][r] + d[2 * c + 1][r] * d[2 * c + 1][r];
            // 4-stage butterfly WHT over n = lane&15 (basis index)
#pragma unroll
            for (int st = 0; st < 4; ++st) {
                int msk = 1 << st;
                bool up = (lane & msk) != 0;
#pragma unroll
                for (int r = 0; r < 8; ++r) {
                    float wv = __shfl_xor(v[r], msk);
                    v[r] = up ? (wv - v[r]) : (v[r] + wv);
                }
            }
            // z_i lives at single-bit masks: n=8->z0, 4->z1, 2->z2, 1->z3
            int n = lane & 15;
            int qi = (n == 8) ? 0 : (n == 4) ? 1 : (n == 2) ? 2 : (n == 1) ? 3 : -1;
            if (qi >= 0) {
                float* qp = qbuf + (size_t)(row0 + 8 * kh) * 12 + 4 * c + qi;
#pragma unroll
                for (int r = 0; r < 8; ++r)
                    qp[(size_t)r * 12] = v[r];
            }
        }
    }
}

// ---------------------------------------------------------------------------
// K2: deterministic moment accumulation of q: 12 sums + 78 second moments
// ---------------------------------------------------------------------------
__global__ __launch_bounds__(256) void k_stats(const float* __restrict__ qbuf,
                                               float* __restrict__ partial, int B) {
    float s1[12], s2[78];
#pragma unroll
    for (int i = 0; i < 12; ++i) s1[i] = 0.f;
#pragma unroll
    for (int i = 0; i < 78; ++i) s2[i] = 0.f;
    for (int row = blockIdx.x * blockDim.x + threadIdx.x; row < B;
         row += gridDim.x * blockDim.x) {
        const float4* qp = (const float4*)(qbuf + (size_t)row * 12);
        float4 a = qp[0], b = qp[1], c = qp[2];
        float qv[12] = {a.x, a.y, a.z, a.w, b.x, b.y, b.z, b.w, c.x, c.y, c.z, c.w};
#pragma unroll
        for (int i = 0; i < 12; ++i) s1[i] += qv[i];
        int idx = 0;
#pragma unroll
        for (int i = 0; i < 12; ++i)
#pragma unroll
            for (int j = 0; j <= i; ++j) s2[idx++] += qv[i] * qv[j];
    }
    __shared__ float red[8][90];
    int lane = threadIdx.x & 31, w = threadIdx.x >> 5;
#pragma unroll
    for (int v = 0; v < 90; ++v) {
        float t = (v < 12) ? s1[v] : s2[v - 12];
#pragma unroll
        for (int off = 16; off; off >>= 1) t += __shfl_xor(t, off);
        if (lane == 0) red[w][v] = t;
    }
    __syncthreads();
    if (threadIdx.x < 90) {
        float tot = 0.f;
#pragma unroll
        for (int w2 = 0; w2 < 8; ++w2) tot += red[w2][threadIdx.x];
        partial[blockIdx.x * 90 + threadIdx.x] = tot;
    }
}

// ---------------------------------------------------------------------------
// K3: fold q-moments + fc_w + gamma/beta into per-feature scale/shift.
// var_h[j] = w_j^T Cov(q) w_j ; bias cancels in (h - mu).
// ---------------------------------------------------------------------------
__global__ void k_finalize(const float* __restrict__ partial, int nblk,
                           const float* __restrict__ fcw, const float* __restrict__ fcb,
                           const float* __restrict__ gamma, const float* __restrict__ beta,
                           float* __restrict__ scsh, float Binv) {
    __shared__ float S[90];
    __shared__ float mq[12];
    __shared__ float cov[12][12];
    int t = threadIdx.x;
    (void)fcb;
    if (t < 90) {
        float s = 0.f;
        for (int b = 0; b < nblk; ++b) s += partial[b * 90 + t];
        S[t] = s;
    }
    __syncthreads();
    if (t < 12) mq[t] = S[t] * Binv;
    __syncthreads();
    if (t < 78) {
        int i = 0;
        while ((i + 1) * (i + 2) / 2 <= t) ++i;
        int j = t - i * (i + 1) / 2;
        float c = S[12 + t] * Binv - mq[i] * mq[j];
        cov[i][j] = c; cov[j][i] = c;
    }
    __syncthreads();
    if (t < 128) {
        float wv[12];
#pragma unroll
        for (int i = 0; i < 12; ++i) wv[i] = fcw[t * 12 + i];
        float mu = 0.f;
#pragma unroll
        for (int i = 0; i < 12; ++i) mu += wv[i] * mq[i];
        float var = 0.f;
#pragma unroll
        for (int i = 0; i < 12; ++i)
#pragma unroll
            for (int j = 0; j < 12; ++j) var += wv[i] * wv[j] * cov[i][j];
        float scale = gamma[t] * rsqrtf(var + 1e-5f);
        scsh[t] = scale;
        scsh[128 + t] = beta[t] - mu * scale;
    }
}

// ---------------------------------------------------------------------------
// K4: fused output pass: h = q*W^T (24x WMMA), h*scale+shift, ReLU, store.
// ---------------------------------------------------------------------------
__global__ __launch_bounds__(256) void k_pass2(const float* __restrict__ qbuf,
                                               const float* __restrict__ fcw,
                                               const float* __restrict__ scsh,
                                               float* __restrict__ out, int ntiles) {
    const int lane = threadIdx.x & 31;
    const int m = lane & 15;
    const int kh = lane >> 4;
    const int wid = blockIdx.x * (blockDim.x >> 5) + (threadIdx.x >> 5);
    const int nw = gridDim.x * (blockDim.x >> 5);

    v2f wb[8][3];
    float sc[8], sh[8];
#pragma unroll
    for (int t = 0; t < 8; ++t) {
#pragma unroll
        for (int ks = 0; ks < 3; ++ks)
            wb[t][ks] = *(const v2f*)(fcw + (16 * t + m) * 12 + 4 * ks + 2 * kh);
        sc[t] = scsh[16 * t + m];
        sh[t] = scsh[128 + 16 * t + m];
    }
    for (int tile = wid; tile < ntiles; tile += nw) {
        const int row0 = tile << 4;
        v2f af[3];
#pragma unroll
        for (int ks = 0; ks < 3; ++ks)
            af[ks] = *(const v2f*)(qbuf + (size_t)(row0 + m) * 12 + 4 * ks + 2 * kh);
#pragma unroll
        for (int t = 0; t < 8; ++t) {
            v8f acc = {};
#pragma unroll
            for (int ks = 0; ks < 3; ++ks)
                acc = wmma4(af[ks], wb[t][ks], acc);
#pragma unroll
            for (int r = 0; r < 8; ++r) {
                float h = fmaxf(acc[r] * sc[t] + sh[t], 0.f);
                out[(size_t)(row0 + r + 8 * kh) * 128 + 16 * t + m] = h;
            }
        }
    }
}

// ---------------------------------------------------------------------------
extern "C" void kernel_launch(void* const* d_in, const int* in_sizes, int n_in,
                              void* d_out, int out_size, void* d_ws, size_t ws_size,
                              hipStream_t stream) {
    (void)n_in; (void)out_size; (void)ws_size;
    const float* x     = (const float*)d_in[0];
    const float* p1    = (const float*)d_in[1];
    const float* p2    = (const float*)d_in[2];
    const float* p3    = (const float*)d_in[3];
    const float* fcw   = (const float*)d_in[4];
    const float* fcb   = (const float*)d_in[5];
    const float* gamma = (const float*)d_in[6];
    const float* beta  = (const float*)d_in[7];
    float* out = (float*)d_out;

    const int B = in_sizes[0] / 16;
    const int ntiles = B / 16;

    float* ws      = (float*)d_ws;
    float* Amat    = ws;               // 96*16 = 1536 floats
    float* scsh    = ws + 1536;        // 256 floats (scale[128], shift[128])
    float* partial = ws + 1792;        // 256 blocks * 90 = 23040 floats
    float* qbuf    = ws + 24832;       // B*12 floats (16B-aligned offset)

    const int SB = 256;

    hipLaunchKernelGGL(k_build, dim3(1), dim3(64), 0, stream, p1, p2, p3, Amat);
    hipLaunchKernelGGL(k_pass1, dim3(1024), dim3(256), 0, stream, x, Amat, qbuf, ntiles);
    hipLaunchKernelGGL(k_stats, dim3(SB), dim3(256), 0, stream, qbuf, partial, B);
    hipLaunchKernelGGL(k_finalize, dim3(1), dim3(128), 0, stream,
                       partial, SB, fcw, fcb, gamma, beta, scsh, 1.0f / (float)B);
    hipLaunchKernelGGL(k_pass2, dim3(1024), dim3(256), 0, stream, qbuf, fcw, scsh, out, ntiles);
}